// SparseAttentionHead_893353198191
// MI455X (gfx1250) — compile-verified
//
#include <hip/hip_runtime.h>
#include <hip/hip_bf16.h>
#include <math.h>

typedef __attribute__((ext_vector_type(2))) float v2f;
typedef __attribute__((ext_vector_type(8))) float v8f;

#define NN   8192
#define CH   128
#define TJ   64     // j-columns per chunk
#define RPB  32     // rows per block
#define T2   256    // threads per block (8 wave32)
#define PLSTR 68    // padded LDS row stride (floats): conflict-free WMMA-A reads

// ---------------------------------------------------------------------------
// Kernel 1: seq_fts = W1 @ x  (128 x 8192), stored pre-swizzled into the exact
// per-lane WMMA-B operand order ("SB"), plus attention scalars a1[n], a2[n].
//
// B-operand layout for V_WMMA_F32_16X16X4_F32 (4x16, K x N):
//   lane l (l%16 = column n, l/16 = h), VGPR v holds K = v + 2*h.
// SB linear index for tile (kt = j/4, ct = c/16):
//   SB[(kt*8 + ct)*64 + (h*16 + cc)*2 + v]   with j = 4*kt + (2*h + v), c = 16*ct + cc
// ---------------------------------------------------------------------------
__global__ __launch_bounds__(128) void gat_prep(
    const float* __restrict__ x,   // (128, 8192)
    const float* __restrict__ W1,  // (128, 128)
    const float* __restrict__ wf1, const float* __restrict__ bf1,
    const float* __restrict__ wf2, const float* __restrict__ bf2,
    float* __restrict__ SB,        // 8192*128 floats, swizzled
    float* __restrict__ a1,        // 8192
    float* __restrict__ a2)        // 8192
{
  __shared__ float xs[128][17];   // padded: conflict-free column reductions
  const int t  = threadIdx.x;     // 0..127 -> output channel c
  const int j0 = blockIdx.x * 16; // 16 nodes per block

  for (int i = t; i < 128 * 16; i += 128) {
    int c = i >> 4, n = i & 15;
    xs[c][n] = x[c * NN + j0 + n];
  }
  __syncthreads();

  const int c = t;
  float acc[16];
#pragma unroll
  for (int n = 0; n < 16; ++n) acc[n] = 0.f;

  for (int k = 0; k < 128; ++k) {
    float w = W1[c * 128 + k];
#pragma unroll
    for (int n = 0; n < 16; ++n) acc[n] = fmaf(w, xs[k][n], acc[n]);
  }

  // scatter into WMMA-B swizzled layout
  const int ct = c >> 4, cc = c & 15;
#pragma unroll
  for (int n = 0; n < 16; ++n) {
    int j  = j0 + n;
    int kt = j >> 2, kk = j & 3;
    int h  = kk >> 1, v = kk & 1;
    SB[(size_t)(kt * 8 + ct) * 64 + (h * 16 + cc) * 2 + v] = acc[n];
  }

  // a1[j] = sum_c wf1[c]*seq_fts[c][j] + b1   (and same for a2)
  __syncthreads();
#pragma unroll
  for (int n = 0; n < 16; ++n) xs[c][n] = wf1[c] * acc[n];
  __syncthreads();
  if (t < 16) {
    float s = bf1[0];
    for (int c2 = 0; c2 < 128; ++c2) s += xs[c2][t];
    a1[j0 + t] = s;
  }
  __syncthreads();
#pragma unroll
  for (int n = 0; n < 16; ++n) xs[c][n] = wf2[c] * acc[n];
  __syncthreads();
  if (t < 16) {
    float s = bf2[0];
    for (int c2 = 0; c2 < 128; ++c2) s += xs[c2][t];
    a2[j0 + t] = s;
  }
}

// ---------------------------------------------------------------------------
// Kernel 2: fused flash-softmax attention aggregation.
// Each block owns 32 output rows. Stream adj once in 32x64 chunks:
//   phase 1 (all 256 threads): p = exp(leaky_relu(adj*(a1_i+a2_j)) - b_i) -> LDS
//   phase 2 (8 waves): v_wmma_f32_16x16x4_f32 accumulate P x seq_fts^T
// Row sums stay in registers (thread<->row fixed), final scale + ELU on store.
// ---------------------------------------------------------------------------
__global__ __launch_bounds__(T2) void gat_attend(
    const float* __restrict__ adj,  // (8192, 8192)
    const float* __restrict__ SB,   // swizzled seq_fts^T
    const float* __restrict__ a1,
    const float* __restrict__ a2,
    float* __restrict__ out)        // (8192, 128)
{
  __shared__ float Pl[RPB][PLSTR];
  __shared__ float srow[RPB];
  __shared__ float redm[T2];

  const int t  = threadIdx.x;
  const int i0 = blockIdx.x * RPB;

  // ---- block-wide max of a2 -> per-row softmax shift bound b_i ----
  float m = -1e30f;
  for (int j = t; j < NN; j += T2) m = fmaxf(m, a2[j]);
  redm[t] = m;
  __syncthreads();
  for (int s = T2 / 2; s > 0; s >>= 1) {
    if (t < s) redm[t] = fmaxf(redm[t], redm[t + s]);
    __syncthreads();
  }
  const float maxa2 = redm[0];

  // ---- phase-1 mapping: 8 threads per row, 8 columns each ----
  const int row = t >> 3, q = t & 7;
  const float a1r  = a1[i0 + row];
  const float br   = fmaxf(0.f, a1r + maxa2);   // >= every logit in this row
  float       sumr = 0.f;
  const float* arow = adj + (size_t)(i0 + row) * NN + q * 8;
  const float* a2q  = a2 + q * 8;

  // ---- phase-2 mapping: wave w -> row-tile (w&1), c-tiles {2*(w>>1), +1} ----
  const int w = t >> 5, lane = t & 31;
  const int rtile = w & 1, cq = w >> 1;
  const int half = lane >> 4, mrow = lane & 15;
  const int ldsrow = rtile * 16 + mrow;
  const v2f* __restrict__ SB2 = (const v2f*)SB;

  v8f accv[2];
#pragma unroll
  for (int u = 0; u < 2; ++u)
    accv[u] = (v8f){0.f, 0.f, 0.f, 0.f, 0.f, 0.f, 0.f, 0.f};

  for (int j0 = 0; j0 < NN; j0 += TJ) {
    __syncthreads();   // previous chunk's WMMA reads of Pl complete

    if (j0 + TJ < NN) __builtin_prefetch(arow + j0 + TJ, 0, 1);

    // phase 1: compute 8 probabilities per thread, stage to LDS
#pragma unroll
    for (int u2 = 0; u2 < 2; ++u2) {
      float4 av = *(const float4*)(arow + j0 + 4 * u2);
      float4 tv = *(const float4*)(a2q  + j0 + 4 * u2);
      float4 pv;
      {
        float l = av.x * (a1r + tv.x); l = l > 0.f ? l : 0.01f * l;
        pv.x = __expf(l - br);
        l = av.y * (a1r + tv.y); l = l > 0.f ? l : 0.01f * l;
        pv.y = __expf(l - br);
        l = av.z * (a1r + tv.z); l = l > 0.f ? l : 0.01f * l;
        pv.z = __expf(l - br);
        l = av.w * (a1r + tv.w); l = l > 0.f ? l : 0.01f * l;
        pv.w = __expf(l - br);
      }
      sumr += (pv.x + pv.y) + (pv.z + pv.w);
      *(float4*)&Pl[row][q * 8 + 4 * u2] = pv;
    }
    __syncthreads();   // Pl tile ready

    // phase 2: 16 K-steps of fp32 WMMA over this 64-wide chunk
    const int ktb = j0 >> 2;
#pragma unroll 4
    for (int ks = 0; ks < 16; ++ks) {
      const int k0 = ks * 4;
      v2f a = *(const v2f*)&Pl[ldsrow][k0 + 2 * half];   // WMMA-A operand
      const int kt = ktb + ks;
#pragma unroll
      for (int u = 0; u < 2; ++u) {
        v2f b = SB2[(size_t)(kt * 8 + (2 * cq + u)) * 32 + lane];
        accv[u] = __builtin_amdgcn_wmma_f32_16x16x4_f32(
            false, a, false, b, (short)0, accv[u], false, false);
      }
    }
  }

  // ---- finalize: row-sum reduce (8 lanes/row, consecutive) ----
  sumr += __shfl_xor(sumr, 1, 32);
  sumr += __shfl_xor(sumr, 2, 32);
  sumr += __shfl_xor(sumr, 4, 32);
  if (q == 0) srow[row] = sumr;
  __syncthreads();

  // C/D layout: VGPR r, lane l -> row r + 8*(l/16), col l%16
#pragma unroll
  for (int r = 0; r < 8; ++r) {
    const int rl  = rtile * 16 + r + 8 * half;
    const float inv = 1.0f / srow[rl];
#pragma unroll
    for (int u = 0; u < 2; ++u) {
      float v = accv[u][r] * inv;
      v = v > 0.f ? v : (__expf(v) - 1.0f);          // ELU(alpha=1)
      out[(size_t)(i0 + rl) * CH + (2 * cq + u) * 16 + mrow] = v;
    }
  }
}

// ---------------------------------------------------------------------------
extern "C" void kernel_launch(void* const* d_in, const int* in_sizes, int n_in,
                              void* d_out, int out_size, void* d_ws, size_t ws_size,
                              hipStream_t stream) {
  const float* x   = (const float*)d_in[0];
  // d_in[1] = edge_index (int64) -- unused by the reference computation
  const float* adj = (const float*)d_in[2];
  const float* W1  = (const float*)d_in[3];
  const float* wf1 = (const float*)d_in[4];
  const float* bf1 = (const float*)d_in[5];
  const float* wf2 = (const float*)d_in[6];
  const float* bf2 = (const float*)d_in[7];
  float* out = (float*)d_out;

  float* SB = (float*)d_ws;                 // 8192*128 floats = 4 MB
  float* a1 = SB + (size_t)NN * CH;         // 8192 floats
  float* a2 = a1 + NN;                      // 8192 floats

  gat_prep<<<NN / 16, 128, 0, stream>>>(x, W1, wf1, bf1, wf2, bf2, SB, a1, a2);
  gat_attend<<<NN / RPB, T2, 0, stream>>>(adj, SB, a1, a2, out);
}